// CapsuleLayer_3616362463961
// MI455X (gfx1250) — compile-verified
//
#include <hip/hip_runtime.h>
#include <math.h>

typedef __attribute__((ext_vector_type(2))) float v2f;
typedef __attribute__((ext_vector_type(8))) float v8f;

#define NCAPS   10
#define BTOT    256
#define NROUTE  1152
#define CIN     8
#define COUT    16
#define BT      16
#define NWAVES  8
#define THREADS 256
#define LPITCH  (NROUTE + 1)   // 1153: 1153 % 64 == 1 -> conflict-free banks

// Compute (or accumulate onto `cin`) the [16b x 16o] prior tile for route node n:
// prior[b][o] = sum_i x[b0+b][n][i] * w[c][n][i][o], K=8 via two WMMA f32 16x16x4.
// A layout: lane l<16 holds A[l][2h],A[l][2h+1] (h = lane>=16) per ISA 16x4 f32 spec.
// B layout: lane l<16 holds B[2h][l],B[2h+1][l].
__device__ __forceinline__ v8f prior_acc(const float* __restrict__ x,
                                         const float* __restrict__ w,
                                         int c, int b0, int n, int l, int hf,
                                         v8f cin) {
    const float* xp = x + ((size_t)(b0 + l) * NROUTE + n) * CIN + hf * 2;
    v2f a0, a1, bb0, bb1;
    a0.x = xp[0]; a0.y = xp[1];          // k = 2h, 2h+1
    a1.x = xp[4]; a1.y = xp[5];          // k = 2h+4, 2h+5
    const float* wp = w + (((size_t)c * NROUTE + n) * CIN + hf * 2) * COUT + l;
    bb0.x = wp[0];        bb0.y = wp[COUT];        // rows 2h, 2h+1 at col l
    bb1.x = wp[4 * COUT]; bb1.y = wp[5 * COUT];    // rows 2h+4, 2h+5
    v8f acc = cin;
    acc = __builtin_amdgcn_wmma_f32_16x16x4_f32(false, a0, false, bb0, (short)0, acc, false, false);
    acc = __builtin_amdgcn_wmma_f32_16x16x4_f32(false, a1, false, bb1, (short)0, acc, false, false);
    return acc;
}

// Cross-wave reduce s partials, then squash -> l_out[b*16+o]. Called by all 256 threads.
__device__ __forceinline__ void reduce_squash(float* __restrict__ l_red,
                                              float* __restrict__ l_s,
                                              float* __restrict__ l_scale,
                                              float* __restrict__ l_out,
                                              const v8f& acc, int t, int wv, int lane,
                                              float sfac) {
#pragma unroll
    for (int v = 0; v < 8; ++v) l_red[wv * 256 + v * 32 + lane] = acc[v];
    __syncthreads();
    float sval = 0.f;
#pragma unroll
    for (int w2 = 0; w2 < NWAVES; ++w2) sval += l_red[w2 * 256 + t];
    // element (v = t>>5, lane = t&31) maps to b = v + 8*(lane>=16), o = lane&15
    int b = (t >> 5) + (((t & 31) >= 16) ? 8 : 0);
    int o = t & 15;
    l_s[b * 16 + o] = sval * sfac;
    __syncthreads();
    if (t < 16) {
        float sq = 0.f;
        for (int oo = 0; oo < 16; ++oo) { float sv = l_s[t * 16 + oo]; sq += sv * sv; }
        l_scale[t] = sqrtf(sq) / (1.0f + sq);   // squash: s * sqrt(sq)/(1+sq)
    }
    __syncthreads();
    l_out[t] = l_s[t] * l_scale[t >> 4];
    __syncthreads();
}

__global__ void __launch_bounds__(THREADS)
capsule_route_kernel(const float* __restrict__ x, const float* __restrict__ w,
                     float* __restrict__ out) {
    __shared__ float l_logits[BT][LPITCH];   // 73.8 KB: logits[b][n]
    __shared__ float l_red[NWAVES * 256];    // 8 KB: wave partials / softmax scratch
    __shared__ float l_s[BT * COUT];
    __shared__ float l_out[BT * COUT];
    __shared__ float l_scale[BT];
    __shared__ float l_max[BT];
    __shared__ float l_invden[BT];

    const int t    = threadIdx.x;
    const int lane = t & 31;
    const int wv   = t >> 5;
    const int l    = lane & 15;
    const int hf   = lane >> 4;
    const int b0   = blockIdx.x * BT;
    const int c    = blockIdx.y;

    // zero logits
    float* lf = &l_logits[0][0];
    for (int j = t; j < BT * LPITCH; j += THREADS) lf[j] = 0.f;
    __syncthreads();

    // ---- Pass 0: uniform probs (softmax of zeros) -> s0 = mean_n prior ----
    {
        v8f acc = {};
        for (int n = wv; n < NROUTE; n += NWAVES)
            acc = prior_acc(x, w, c, b0, n, l, hf, acc);   // pure WMMA accumulation chain
        reduce_squash(l_red, l_s, l_scale, l_out, acc, t, wv, lane, 1.0f / (float)NROUTE);
    }

    // ---- Routing iterations 1..2 ----
    for (int it = 0; it < 2; ++it) {
        // logits[b][n] += sum_o prior[n][b][o] * out[b][o]
        for (int n = wv; n < NROUTE; n += NWAVES) {
            v8f z = {};
            v8f p = prior_acc(x, w, c, b0, n, l, hf, z);
#pragma unroll
            for (int v = 0; v < 8; ++v) {
                int b = hf * 8 + v;
                float tv = p[v] * l_out[b * 16 + l];
                tv += __shfl_xor(tv, 1, 32);
                tv += __shfl_xor(tv, 2, 32);
                tv += __shfl_xor(tv, 4, 32);
                tv += __shfl_xor(tv, 8, 32);
                if (l == v) l_logits[b][n] += tv;   // one lane per half writes
            }
        }
        __syncthreads();

        // softmax stats over n per b: max and inverse denominator
        {
            const int CH = NROUTE / 16;   // 72
            int b = t & 15, ch = t >> 4;
            float m = -__builtin_inff();
            for (int j = 0; j < CH; ++j) m = fmaxf(m, l_logits[b][ch * CH + j]);
            l_red[ch * 16 + b] = m;
            __syncthreads();
            if (t < 16) {
                float mm = -__builtin_inff();
                for (int k2 = 0; k2 < 16; ++k2) mm = fmaxf(mm, l_red[k2 * 16 + t]);
                l_max[t] = mm;
            }
            __syncthreads();
            float mb = l_max[b], ss = 0.f;
            for (int j = 0; j < CH; ++j) ss += __expf(l_logits[b][ch * CH + j] - mb);
            l_red[ch * 16 + b] = ss;
            __syncthreads();
            if (t < 16) {
                float d = 0.f;
                for (int k2 = 0; k2 < 16; ++k2) d += l_red[k2 * 16 + t];
                l_invden[t] = 1.0f / d;
            }
            __syncthreads();
        }

        // weighted s: s[b][o] = sum_n prob[b][n] * prior[n][b][o]
        v8f acc = {};
        for (int n = wv; n < NROUTE; n += NWAVES) {
            v8f z = {};
            v8f p = prior_acc(x, w, c, b0, n, l, hf, z);
#pragma unroll
            for (int v = 0; v < 8; ++v) {
                int b = hf * 8 + v;
                float prob = __expf(l_logits[b][n] - l_max[b]) * l_invden[b];
                acc[v] += prob * p[v];
            }
        }
        reduce_squash(l_red, l_s, l_scale, l_out, acc, t, wv, lane, 1.0f);
    }

    // write final outputs: [C, B, 1, 1, O]
    int b = t >> 4, o = t & 15;
    out[((size_t)c * BTOT + (b0 + b)) * COUT + o] = l_out[t];
}

extern "C" void kernel_launch(void* const* d_in, const int* in_sizes, int n_in,
                              void* d_out, int out_size, void* d_ws, size_t ws_size,
                              hipStream_t stream) {
    (void)in_sizes; (void)n_in; (void)d_ws; (void)ws_size; (void)out_size;
    const float* x = (const float*)d_in[0];   // [256,1152,8]
    const float* w = (const float*)d_in[1];   // [10,1152,8,16]
    float* out = (float*)d_out;               // [10,256,1,1,16]
    dim3 grid(BTOT / BT, NCAPS, 1);
    capsule_route_kernel<<<grid, THREADS, 0, stream>>>(x, w, out);
}